// StockPredictor_69870527971442
// MI455X (gfx1250) — compile-verified
//
#include <hip/hip_runtime.h>

#define D_    128
#define M_    10
#define H_    4
#define J_    32
#define T_    5
#define B_    256
#define S_    1024
#define F_    32
#define DH_   32
#define NROWS (B_ * S_)   // 262144

typedef __attribute__((ext_vector_type(2))) float v2f;
typedef __attribute__((ext_vector_type(8))) float v8f;
typedef int v4i_vs __attribute__((vector_size(16)));   // 16B payload type for async LDS copy

#if defined(__gfx1250__) && __has_builtin(__builtin_amdgcn_global_load_async_to_lds_b128)
#define HAVE_ASYNC_LDS 1
#else
#define HAVE_ASYNC_LDS 0
#endif

#if HAVE_ASYNC_LDS
// Direct memory->LDS copy, 16B per lane, tracked by ASYNCcnt.
__device__ __forceinline__ void async_g2l_b128(const void* g, void* l) {
  __builtin_amdgcn_global_load_async_to_lds_b128(
      (__attribute__((address_space(1))) v4i_vs*)g,
      (__attribute__((address_space(3))) v4i_vs*)l, 0, 0);
}
__device__ __forceinline__ void wait_async0() {
#if __has_builtin(__builtin_amdgcn_s_wait_asynccnt)
  __builtin_amdgcn_s_wait_asynccnt(0);
#else
  asm volatile("s_wait_asynccnt 0x0" ::: "memory");
#endif
}
#endif

__device__ __forceinline__ v8f wmma_f32_k4(v2f a, v2f b, v8f c) {
  // D = A(16x4) * B(4x16) + C, fp32, exact wrt reference math
  return __builtin_amdgcn_wmma_f32_16x16x4_f32(
      /*neg_a=*/false, a, /*neg_b=*/false, b,
      /*c_mod=*/(short)0, c, /*reuse_a=*/false, /*reuse_b=*/false);
}

// ---------------------------------------------------------------------------
// Kernel 0: initialize recurrent state (z, hist, traces) each launch.
// ---------------------------------------------------------------------------
__global__ void init_state_kernel(const float* __restrict__ z_init,
                                  const float* __restrict__ hist_init,
                                  float* __restrict__ z, float* __restrict__ hist,
                                  float* __restrict__ tra, float* __restrict__ tro) {
  int i = blockIdx.x * blockDim.x + threadIdx.x;
  if (i < B_ * D_ * M_) hist[i] = hist_init[i % (D_ * M_)];
  if (i < B_ * D_)      z[i]    = z_init[i % D_];
  if (i < B_ * J_ * 2)  { tra[i] = 0.f; tro[i] = 0.f; }
}

// ---------------------------------------------------------------------------
// Kernel 1: fused  kv = x*W_feat^T + b  ->  K = kv*Wk^T + bk, V = kv*Wv^T + bv
// One workgroup (4 waves) handles a 16-row tile; kv stays in LDS.
// x tile arrives via GLOBAL_LOAD_ASYNC_TO_LDS_B128 (ASYNCcnt); matmuls on
// v_wmma_f32_16x16x4_f32.
// ---------------------------------------------------------------------------
__global__ __launch_bounds__(128) void proj_kv_kernel(
    const float* __restrict__ x, const float* __restrict__ W_feat,
    const float* __restrict__ b_feat, const float* __restrict__ in_proj_w,
    const float* __restrict__ in_proj_b,
    float* __restrict__ kbuf, float* __restrict__ vbuf) {
  __shared__ float xt[16][F_ + 2];    // 16x32 x-tile (padded)
  __shared__ float kvt[16][D_ + 4];   // 16x128 kv-tile (padded)

  const int tid  = threadIdx.x;
  const int lane = tid & 31;
  const int wv   = tid >> 5;
  const int row0 = blockIdx.x * 16;

#if HAVE_ASYNC_LDS
  {
    // 16x32 x-tile is one contiguous 2KB block: 128 lanes x 16B async copies.
    const int e = tid * 4;                       // first of 4 floats, stays in-row
    async_g2l_b128(x + (size_t)row0 * F_ + e, &xt[e >> 5][e & 31]);
    wait_async0();
  }
#else
  for (int i = tid; i < 16 * F_; i += 128) {
    int r = i >> 5, c = i & 31;
    xt[r][c] = x[(size_t)(row0 + r) * F_ + c];
  }
#endif
  __syncthreads();

  const int m     = lane & 15;        // M row (A/C) == N col (B/C) for this lane
  const int khalf = (lane >> 4) * 2;  // K sub-offset: lanes 0-15 -> {0,1}, 16-31 -> {2,3}
  const int mbase = (lane >> 4) * 8;  // C/D rows: VGPR r -> M=r or r+8

  // Phase A: kv = x @ W_feat^T + b_feat      (K = F = 32)
  for (int ct = 0; ct < 2; ++ct) {
    const int ncol = wv * 32 + ct * 16 + m;
    v8f acc = {};
    for (int kk = 0; kk < F_; kk += 4) {
      v2f a, b;
      a.x = xt[m][kk + khalf];
      a.y = xt[m][kk + khalf + 1];
      b.x = W_feat[ncol * F_ + kk + khalf];
      b.y = W_feat[ncol * F_ + kk + khalf + 1];
      acc = wmma_f32_k4(a, b, acc);
    }
    const float bias = b_feat[ncol];
#pragma unroll
    for (int r = 0; r < 8; ++r) kvt[mbase + r][ncol] = acc[r] + bias;
  }
  __syncthreads();

  // Phase B: K = kv @ Wk^T + bk ; V = kv @ Wv^T + bv   (K = D = 128)
  for (int mat = 0; mat < 2; ++mat) {
    const float* Wm = in_proj_w + (size_t)(D_ + mat * D_) * D_;  // Wk then Wv rows
    const float* bm = in_proj_b + D_ + mat * D_;
    float* out = (mat == 0) ? kbuf : vbuf;
    for (int ct = 0; ct < 2; ++ct) {
      const int ncol = wv * 32 + ct * 16 + m;
      v8f acc = {};
      for (int kk = 0; kk < D_; kk += 4) {
        v2f a, b;
        a.x = kvt[m][kk + khalf];
        a.y = kvt[m][kk + khalf + 1];
        b.x = Wm[ncol * D_ + kk + khalf];
        b.y = Wm[ncol * D_ + kk + khalf + 1];
        acc = wmma_f32_k4(a, b, acc);
      }
      const float bias = bm[ncol];
#pragma unroll
      for (int r = 0; r < 8; ++r)
        out[(size_t)(row0 + mbase + r) * D_ + ncol] = acc[r] + bias;
    }
  }
}

// ---------------------------------------------------------------------------
// Kernel 2 (per tick): synch traces, mean/sigma heads, q = act->Wq projection.
// One block per batch element; tiny latency-bound kernel.
// ---------------------------------------------------------------------------
__global__ __launch_bounds__(128) void synch_q_kernel(
    const float* __restrict__ z, float* __restrict__ tra, float* __restrict__ tro,
    const float* __restrict__ r_action, const float* __restrict__ r_out,
    const int* __restrict__ ia_act, const int* __restrict__ ja_act,
    const int* __restrict__ ia_out, const int* __restrict__ ja_out,
    const float* __restrict__ w_act_w, const float* __restrict__ w_act_b,
    const float* __restrict__ w_mean_w, const float* __restrict__ w_mean_b,
    const float* __restrict__ w_ls_w, const float* __restrict__ w_ls_b,
    const float* __restrict__ in_proj_w, const float* __restrict__ in_proj_b,
    float* __restrict__ qh, float* __restrict__ means, float* __restrict__ sig,
    int t) {
  __shared__ float zsh[D_], s_act[J_], s_out[J_], qsh[D_];
  const int b = blockIdx.x, d = threadIdx.x;
  zsh[d] = z[b * D_ + d];
  __syncthreads();

  if (d < J_) {
    const int j = d;
    float decay = expf(-fmaxf(r_action[j], 0.f));
    float a  = decay * tra[(b * J_ + j) * 2 + 0] + zsh[ia_act[j]] * zsh[ja_act[j]];
    float be = decay * tra[(b * J_ + j) * 2 + 1] + 1.f;
    tra[(b * J_ + j) * 2 + 0] = a;
    tra[(b * J_ + j) * 2 + 1] = be;
    s_act[j] = a * rsqrtf(be + 1e-6f);
  } else if (d < 2 * J_) {
    const int j = d - J_;
    float decay = expf(-fmaxf(r_out[j], 0.f));
    float a  = decay * tro[(b * J_ + j) * 2 + 0] + zsh[ia_out[j]] * zsh[ja_out[j]];
    float be = decay * tro[(b * J_ + j) * 2 + 1] + 1.f;
    tro[(b * J_ + j) * 2 + 0] = a;
    tro[(b * J_ + j) * 2 + 1] = be;
    s_out[j] = a * rsqrtf(be + 1e-6f);
  }
  __syncthreads();

  if (d == 0) {
    float acc = w_mean_b[0];
    for (int j = 0; j < J_; ++j) acc += s_out[j] * w_mean_w[j];
    means[b * T_ + t] = acc;
  } else if (d == 1) {
    float acc = w_ls_b[0];
    for (int j = 0; j < J_; ++j) acc += s_out[j] * w_ls_w[j];
    sig[b * T_ + t] = expf(acc);
  }

  float accq = w_act_b[d];
  for (int j = 0; j < J_; ++j) accq += s_act[j] * w_act_w[d * J_ + j];
  qsh[d] = accq;
  __syncthreads();

  float acch = in_proj_b[d];                    // Wq = in_proj_w rows [0,128)
  for (int i = 0; i < D_; ++i) acch += qsh[i] * in_proj_w[d * D_ + i];
  qh[b * D_ + d] = acch;
}

// ---------------------------------------------------------------------------
// Kernel 3 (per tick): q-len-1 attention over S=1024. One block per (b,h).
// Streams K then V (coalesced 128B rows); V prefetched during logit pass.
// ---------------------------------------------------------------------------
__global__ __launch_bounds__(256) void attn_kernel(
    const float* __restrict__ qh, const float* __restrict__ kbuf,
    const float* __restrict__ vbuf, float* __restrict__ ovec) {
  __shared__ float qsh[DH_];
  __shared__ float lsh[S_];
  __shared__ float red[256];
  __shared__ float part[8][DH_];

  const int b = blockIdx.x >> 2;
  const int h = blockIdx.x & (H_ - 1);
  const int tid = threadIdx.x;
  if (tid < DH_) qsh[tid] = qh[b * D_ + h * DH_ + tid];
  __syncthreads();

  const float scale = 0.17677669529663687f;  // 1/sqrt(32)
  const float* kb = kbuf + (size_t)b * S_ * D_ + h * DH_;
  const float* vb = vbuf + (size_t)b * S_ * D_ + h * DH_;

  float myl[4];
  float lmax = -3.402823466e38f;
  for (int c = 0; c < 4; ++c) {
    const int s = tid + c * 256;
    const float* kr = kb + (size_t)s * D_;
    __builtin_prefetch(vb + (size_t)s * D_, 0, 1);  // global_prefetch_b8 for V stream
    float acc = 0.f;
#pragma unroll
    for (int dd = 0; dd < DH_; ++dd) acc += qsh[dd] * kr[dd];
    acc *= scale;
    myl[c] = acc;
    lmax = fmaxf(lmax, acc);
  }
  red[tid] = lmax;
  __syncthreads();
  for (int off = 128; off > 0; off >>= 1) {
    if (tid < off) red[tid] = fmaxf(red[tid], red[tid + off]);
    __syncthreads();
  }
  const float gmax = red[0];
  __syncthreads();

  float lsum = 0.f;
  for (int c = 0; c < 4; ++c) {
    const int s = tid + c * 256;
    const float p = expf(myl[c] - gmax);
    lsh[s] = p;
    lsum += p;
  }
  red[tid] = lsum;
  __syncthreads();
  for (int off = 128; off > 0; off >>= 1) {
    if (tid < off) red[tid] += red[tid + off];
    __syncthreads();
  }
  const float inv = 1.f / red[0];

  const int dd = tid & 31, chunk = tid >> 5;
  float acc = 0.f;
  for (int s = chunk * 128; s < chunk * 128 + 128; ++s)
    acc += lsh[s] * vb[(size_t)s * D_ + dd];
  part[chunk][dd] = acc;
  __syncthreads();
  if (chunk == 0) {
    float o = 0.f;
#pragma unroll
    for (int c2 = 0; c2 < 8; ++c2) o += part[c2][dd];
    ovec[b * D_ + h * DH_ + dd] = o * inv;
  }
}

// ---------------------------------------------------------------------------
// Kernel 4 (per tick): out_proj -> synapse MLP (LN+ReLU) -> history shift ->
// per-neuron MLP -> new z. One block per batch element, thread = neuron d.
// ---------------------------------------------------------------------------
__global__ __launch_bounds__(128) void syn_nlm_kernel(
    const float* __restrict__ ovec, const float* __restrict__ out_proj_w,
    const float* __restrict__ out_proj_b, float* __restrict__ z,
    const float* __restrict__ syn_w1, const float* __restrict__ syn_b1,
    const float* __restrict__ ln_g, const float* __restrict__ ln_b,
    const float* __restrict__ syn_w2, const float* __restrict__ syn_b2,
    const float* __restrict__ nlm_w1, const float* __restrict__ nlm_b1,
    const float* __restrict__ nlm_w2, const float* __restrict__ nlm_b2,
    float* __restrict__ hist) {
  __shared__ float osh[D_], zsh[D_], ash[D_], hsh[D_], red[D_];
  const int b = blockIdx.x, d = threadIdx.x;
  osh[d] = ovec[b * D_ + d];
  zsh[d] = z[b * D_ + d];
  __syncthreads();

  float ao = out_proj_b[d];
  for (int i = 0; i < D_; ++i) ao += osh[i] * out_proj_w[d * D_ + i];
  ash[d] = ao;
  __syncthreads();

  float hv = syn_b1[d];
  const float* w1 = syn_w1 + (size_t)d * 2 * D_;
  for (int i = 0; i < D_; ++i) hv += zsh[i] * w1[i];
  for (int i = 0; i < D_; ++i) hv += ash[i] * w1[D_ + i];

  red[d] = hv;
  __syncthreads();
  for (int off = 64; off > 0; off >>= 1) {
    if (d < off) red[d] += red[d + off];
    __syncthreads();
  }
  const float mu = red[0] * (1.f / 128.f);
  __syncthreads();
  const float df = hv - mu;
  red[d] = df * df;
  __syncthreads();
  for (int off = 64; off > 0; off >>= 1) {
    if (d < off) red[d] += red[d + off];
    __syncthreads();
  }
  const float var = red[0] * (1.f / 128.f);
  float hn = df * rsqrtf(var + 1e-5f) * ln_g[d] + ln_b[d];
  hsh[d] = fmaxf(hn, 0.f);
  __syncthreads();

  float an = syn_b2[d];
  for (int i = 0; i < D_; ++i) an += hsh[i] * syn_w2[d * D_ + i];

  float hl[M_];
  float* hr = hist + ((size_t)b * D_ + d) * M_;
#pragma unroll
  for (int mm = 0; mm < M_ - 1; ++mm) hl[mm] = hr[mm + 1];
  hl[M_ - 1] = an;
#pragma unroll
  for (int mm = 0; mm < M_; ++mm) hr[mm] = hl[mm];

  const float* wa = nlm_w1 + (size_t)d * M_ * 32;
  const float* ba = nlm_b1 + d * 32;
  float zacc = nlm_b2[d];
  for (int o = 0; o < 32; ++o) {
    float t2 = ba[o];
#pragma unroll
    for (int mm = 0; mm < M_; ++mm) t2 += hl[mm] * wa[mm * 32 + o];
    zacc += fmaxf(t2, 0.f) * nlm_w2[d * 32 + o];
  }
  z[b * D_ + d] = zacc;
}

// ---------------------------------------------------------------------------
// Kernel 5: argmin(sigma over ticks) -> select mean. First-occurrence min.
// ---------------------------------------------------------------------------
__global__ void finalize_kernel(const float* __restrict__ means,
                                const float* __restrict__ sig,
                                float* __restrict__ out) {
  int b = blockIdx.x * blockDim.x + threadIdx.x;
  if (b >= B_) return;
  float best = sig[b * T_];
  int bi = 0;
  for (int t = 1; t < T_; ++t) {
    float s = sig[b * T_ + t];
    if (s < best) { best = s; bi = t; }
  }
  out[b] = means[b * T_ + bi];
}

// ---------------------------------------------------------------------------
extern "C" void kernel_launch(void* const* d_in, const int* in_sizes, int n_in,
                              void* d_out, int out_size, void* d_ws, size_t ws_size,
                              hipStream_t stream) {
  const float* x         = (const float*)d_in[0];
  const float* W_feat    = (const float*)d_in[1];
  const float* b_feat    = (const float*)d_in[2];
  const float* z_init    = (const float*)d_in[3];
  const float* hist_init = (const float*)d_in[4];
  const float* syn_w1    = (const float*)d_in[5];
  const float* syn_b1    = (const float*)d_in[6];
  const float* ln_g      = (const float*)d_in[7];
  const float* ln_b      = (const float*)d_in[8];
  const float* syn_w2    = (const float*)d_in[9];
  const float* syn_b2    = (const float*)d_in[10];
  const float* nlm_w1    = (const float*)d_in[11];
  const float* nlm_b1    = (const float*)d_in[12];
  const float* nlm_w2    = (const float*)d_in[13];
  const float* nlm_b2    = (const float*)d_in[14];
  const float* in_proj_w = (const float*)d_in[15];
  const float* in_proj_b = (const float*)d_in[16];
  const float* out_proj_w= (const float*)d_in[17];
  const float* out_proj_b= (const float*)d_in[18];
  const float* w_mean_w  = (const float*)d_in[19];
  const float* w_mean_b  = (const float*)d_in[20];
  const float* w_ls_w    = (const float*)d_in[21];
  const float* w_ls_b    = (const float*)d_in[22];
  const float* w_act_w   = (const float*)d_in[23];
  const float* w_act_b   = (const float*)d_in[24];
  const float* r_out     = (const float*)d_in[25];
  const float* r_action  = (const float*)d_in[26];
  const int*   idx_out_i = (const int*)d_in[27];
  const int*   idx_out_j = (const int*)d_in[28];
  const int*   idx_act_i = (const int*)d_in[29];
  const int*   idx_act_j = (const int*)d_in[30];

  float* ws = (float*)d_ws;
  size_t off = 0;
  float* kbuf  = ws + off; off += (size_t)NROWS * D_;
  float* vbuf  = ws + off; off += (size_t)NROWS * D_;
  float* z     = ws + off; off += B_ * D_;
  float* hist  = ws + off; off += B_ * D_ * M_;
  float* tra   = ws + off; off += B_ * J_ * 2;
  float* tro   = ws + off; off += B_ * J_ * 2;
  float* qh    = ws + off; off += B_ * D_;
  float* ovec  = ws + off; off += B_ * D_;
  float* means = ws + off; off += B_ * T_;
  float* sig   = ws + off; off += B_ * T_;

  init_state_kernel<<<(B_ * D_ * M_ + 255) / 256, 256, 0, stream>>>(
      z_init, hist_init, z, hist, tra, tro);

  proj_kv_kernel<<<NROWS / 16, 128, 0, stream>>>(
      x, W_feat, b_feat, in_proj_w, in_proj_b, kbuf, vbuf);

  for (int t = 0; t < T_; ++t) {
    synch_q_kernel<<<B_, 128, 0, stream>>>(
        z, tra, tro, r_action, r_out,
        idx_act_i, idx_act_j, idx_out_i, idx_out_j,
        w_act_w, w_act_b, w_mean_w, w_mean_b, w_ls_w, w_ls_b,
        in_proj_w, in_proj_b, qh, means, sig, t);
    attn_kernel<<<B_ * H_, 256, 0, stream>>>(qh, kbuf, vbuf, ovec);
    syn_nlm_kernel<<<B_, 128, 0, stream>>>(
        ovec, out_proj_w, out_proj_b, z,
        syn_w1, syn_b1, ln_g, ln_b, syn_w2, syn_b2,
        nlm_w1, nlm_b1, nlm_w2, nlm_b2, hist);
  }

  finalize_kernel<<<1, 256, 0, stream>>>(means, sig, (float*)d_out);
}